// MultiPersonDecoderPerPerson_28819230556873
// MI455X (gfx1250) — compile-verified
//
#include <hip/hip_runtime.h>
#include <math.h>

// ---- problem constants (match reference) ----
#define BB   32
#define TT   4096
#define DD   256
#define PP   2
#define KQ   17
#define LL   3
#define NHH  8
#define HDD  32
#define FFD  2048
#define HIDD 128

#define NTOK  (BB*PP*KQ)      // 1088  (divisible by 32)
#define NMEM  (BB*TT)         // 131072
#define NPAIR (BB*PP*KQ*KQ)   // 18496 (divisible by 32)

typedef __attribute__((ext_vector_type(2))) float v2f;
typedef __attribute__((ext_vector_type(8))) float v8f;

// ---------------------------------------------------------------------------
// GEMM: C[M,Nc] = A[M,K] @ W[Nw,K]^T (+bias) (+ReLU), fp32-exact via
// V_WMMA_F32_16X16X4_F32.  One wave computes a 32x64 tile of C
// (2 M-subtiles x 4 N-subtiles = 8 independent accumulators).
// Requirements: M % 32 == 0, Nw % 64 == 0, Kd % 4 == 0.  Nc <= Nw is the
// store width / leading dimension (only the final fc uses Nc < Nw).
// ---------------------------------------------------------------------------
template<bool RELU>
__global__ void gemm_wmma_kernel(const float* __restrict__ A,
                                 const float* __restrict__ W,
                                 const float* __restrict__ bias,
                                 float* __restrict__ C,
                                 int M, int Nw, int Nc, int Kd)
{
    const int wave = threadIdx.x >> 5;          // 8 waves per block
    const int lane = threadIdx.x & 31;
    const int tiles_n = Nw >> 6;
    const int tiles_m = M >> 5;
    const int tile = blockIdx.x * 8 + wave;     // wave-uniform
    if (tile >= tiles_m * tiles_n) return;      // uniform branch

    const int tm = (tile / tiles_n) << 5;
    const int tn = (tile % tiles_n) << 6;
    const int half = lane >> 4;                 // 0: lanes 0-15, 1: lanes 16-31
    const int l16  = lane & 15;

    // A fragment rows (two 16-row subtiles); per-lane K offset half*2
    const float* __restrict__ a0 = A + (size_t)(tm + l16) * Kd + half * 2;
    const float* __restrict__ a1 = a0 + (size_t)16 * Kd;
    // B fragment rows: W row n = tn + j*16 + l16
    const float* __restrict__ w0 = W + (size_t)(tn + l16) * Kd + half * 2;
    const size_t ws16 = (size_t)16 * Kd;

    v8f c00 = {0.f,0.f,0.f,0.f,0.f,0.f,0.f,0.f};
    v8f c01 = c00, c02 = c00, c03 = c00;
    v8f c10 = c00, c11 = c00, c12 = c00, c13 = c00;

    #pragma unroll 2
    for (int k = 0; k < Kd; k += 4) {
        const v2f fa0 = *(const v2f*)(a0 + k);
        const v2f fa1 = *(const v2f*)(a1 + k);
        const v2f fb0 = *(const v2f*)(w0 + k);
        const v2f fb1 = *(const v2f*)(w0 + ws16 + k);
        const v2f fb2 = *(const v2f*)(w0 + 2 * ws16 + k);
        const v2f fb3 = *(const v2f*)(w0 + 3 * ws16 + k);
        c00 = __builtin_amdgcn_wmma_f32_16x16x4_f32(false, fa0, false, fb0, (short)0, c00, false, false);
        c01 = __builtin_amdgcn_wmma_f32_16x16x4_f32(false, fa0, false, fb1, (short)0, c01, false, false);
        c02 = __builtin_amdgcn_wmma_f32_16x16x4_f32(false, fa0, false, fb2, (short)0, c02, false, false);
        c03 = __builtin_amdgcn_wmma_f32_16x16x4_f32(false, fa0, false, fb3, (short)0, c03, false, false);
        c10 = __builtin_amdgcn_wmma_f32_16x16x4_f32(false, fa1, false, fb0, (short)0, c10, false, false);
        c11 = __builtin_amdgcn_wmma_f32_16x16x4_f32(false, fa1, false, fb1, (short)0, c11, false, false);
        c12 = __builtin_amdgcn_wmma_f32_16x16x4_f32(false, fa1, false, fb2, (short)0, c12, false, false);
        c13 = __builtin_amdgcn_wmma_f32_16x16x4_f32(false, fa1, false, fb3, (short)0, c13, false, false);
    }

    // C layout: VGPR g -> row (tm + i*16 + g + half*8), col (tn + j*16 + l16)
    v8f acc[2][4] = {{c00, c01, c02, c03}, {c10, c11, c12, c13}};
    #pragma unroll
    for (int i = 0; i < 2; ++i) {
        #pragma unroll
        for (int j = 0; j < 4; ++j) {
            const int col = tn + j * 16 + l16;
            if (col < Nc) {
                const float bv = bias ? bias[col] : 0.f;
                #pragma unroll
                for (int g = 0; g < 8; ++g) {
                    const int row = tm + i * 16 + g + half * 8;
                    float v = acc[i][j][g] + bv;
                    if (RELU) v = fmaxf(v, 0.f);
                    C[(size_t)row * Nc + col] = v;
                }
            }
        }
    }
}

static inline int gemm_blocks(int M, int Nw)
{
    const int tiles = (M >> 5) * (Nw >> 6);
    return (tiles + 7) / 8;
}

// ---------------------------------------------------------------------------
// x init: x[b,p,k,:] = person_q[p] + key_q[k]
// ---------------------------------------------------------------------------
__global__ void init_x_kernel(const float* __restrict__ person_q,
                              const float* __restrict__ key_q,
                              float* __restrict__ x)
{
    const int row = blockIdx.x;                 // 0..NTOK-1
    const int d = threadIdx.x;                  // 0..255
    const int k = row % KQ;
    const int p = (row / KQ) % PP;
    x[(size_t)row * DD + d] = person_q[p * DD + d] + key_q[k * DD + d];
}

// ---------------------------------------------------------------------------
// Self-attention over K=17 keypoint queries. One block per (b,p); one thread
// per (head, query) pair.  qkv rows: [q(256) | k(256) | v(256)].
// ---------------------------------------------------------------------------
__global__ void self_attn_kernel(const float* __restrict__ qkv,
                                 float* __restrict__ out)
{
    const int bp = blockIdx.x;                  // 0..63
    const int pid = threadIdx.x;
    if (pid >= NHH * KQ) return;
    const int h = pid / KQ;
    const int qi = pid % KQ;
    const int base = bp * KQ;
    const float scale = 0.17677669529663687f;   // 1/sqrt(32)

    float qreg[HDD];
    const float* qrow = qkv + (size_t)(base + qi) * (3 * DD) + h * HDD;
    #pragma unroll
    for (int d = 0; d < HDD; ++d) qreg[d] = qrow[d];

    float s[KQ];
    float mx = -1e30f;
    for (int kk = 0; kk < KQ; ++kk) {
        const float* krow = qkv + (size_t)(base + kk) * (3 * DD) + DD + h * HDD;
        float dot = 0.f;
        #pragma unroll
        for (int d = 0; d < HDD; ++d) dot += qreg[d] * krow[d];
        s[kk] = dot * scale;
        mx = fmaxf(mx, s[kk]);
    }
    float sum = 0.f;
    for (int kk = 0; kk < KQ; ++kk) { s[kk] = __expf(s[kk] - mx); sum += s[kk]; }
    const float inv = 1.f / sum;

    for (int d = 0; d < HDD; ++d) {
        float acc = 0.f;
        for (int kk = 0; kk < KQ; ++kk)
            acc += s[kk] * qkv[(size_t)(base + kk) * (3 * DD) + 2 * DD + h * HDD + d];
        out[(size_t)(base + qi) * DD + h * HDD + d] = acc * inv;
    }
}

// ---------------------------------------------------------------------------
// Cross-attention: one block per (b,h,p*q). Scores over T=4096 in LDS.
// ---------------------------------------------------------------------------
__global__ void cross_attn_kernel(const float* __restrict__ Q,   // (NTOK, D)
                                  const float* __restrict__ MK,  // (B*T, D)
                                  const float* __restrict__ MV,  // (B*T, D)
                                  float* __restrict__ O)         // (NTOK, D)
{
    __shared__ float sc[TT];
    __shared__ float red[256];
    __shared__ float qv[HDD];
    __shared__ float part[8][HDD];

    const int blk = blockIdx.x;                 // b*(NH*P*KQ) + h*(P*KQ) + pq
    const int pq = blk % (PP * KQ);
    const int h  = (blk / (PP * KQ)) % NHH;
    const int b  = blk / (PP * KQ * NHH);
    const int tok = b * PP * KQ + pq;
    const int tid = threadIdx.x;
    const float scale = 0.17677669529663687f;

    if (tid < HDD) qv[tid] = Q[(size_t)tok * DD + h * HDD + tid];
    __syncthreads();

    float lmax = -1e30f;
    for (int t = tid; t < TT; t += 256) {
        const float* krow = MK + ((size_t)b * TT + t) * DD + h * HDD;
        float dot = 0.f;
        #pragma unroll
        for (int d = 0; d < HDD; ++d) dot += qv[d] * krow[d];
        dot *= scale;
        sc[t] = dot;
        lmax = fmaxf(lmax, dot);
    }
    red[tid] = lmax; __syncthreads();
    for (int s2 = 128; s2 > 0; s2 >>= 1) {
        if (tid < s2) red[tid] = fmaxf(red[tid], red[tid + s2]);
        __syncthreads();
    }
    const float mx = red[0];
    __syncthreads();

    float lsum = 0.f;
    for (int t = tid; t < TT; t += 256) {
        const float e = __expf(sc[t] - mx);
        sc[t] = e;
        lsum += e;
    }
    red[tid] = lsum; __syncthreads();
    for (int s2 = 128; s2 > 0; s2 >>= 1) {
        if (tid < s2) red[tid] += red[tid + s2];
        __syncthreads();
    }
    const float inv = 1.f / red[0];
    __syncthreads();

    const int d = tid & 31;
    const int chunk = tid >> 5;                 // 8 chunks of 512 keys
    float acc = 0.f;
    for (int t2 = 0; t2 < TT / 8; ++t2) {
        const int t = chunk * (TT / 8) + t2;
        acc += sc[t] * MV[((size_t)b * TT + t) * DD + h * HDD + d];
    }
    part[chunk][d] = acc;
    __syncthreads();
    if (tid < HDD) {
        float o = 0.f;
        #pragma unroll
        for (int c2 = 0; c2 < 8; ++c2) o += part[c2][tid];
        O[(size_t)tok * DD + h * HDD + tid] = o * inv;
    }
}

// ---------------------------------------------------------------------------
// Post-norm: Y = LN(X + R) * w + b.  One block (256 thr) per row, D=256.
// ---------------------------------------------------------------------------
__global__ void ln_resid_kernel(const float* __restrict__ X,
                                const float* __restrict__ R,
                                const float* __restrict__ w,
                                const float* __restrict__ b,
                                float* __restrict__ Y)
{
    __shared__ float red[256];
    const int row = blockIdx.x;
    const int tid = threadIdx.x;
    const float v = X[(size_t)row * DD + tid] + R[(size_t)row * DD + tid];
    red[tid] = v; __syncthreads();
    for (int s2 = 128; s2 > 0; s2 >>= 1) {
        if (tid < s2) red[tid] += red[tid + s2];
        __syncthreads();
    }
    const float mean = red[0] * (1.f / DD);
    __syncthreads();
    const float dv = v - mean;
    red[tid] = dv * dv; __syncthreads();
    for (int s2 = 128; s2 > 0; s2 >>= 1) {
        if (tid < s2) red[tid] += red[tid + s2];
        __syncthreads();
    }
    const float var = red[0] * (1.f / DD);
    Y[(size_t)row * DD + tid] = dv * rsqrtf(var + 1e-5f) * w[tid] + b[tid];
}

// ---------------------------------------------------------------------------
// Refine: build pair[bp,i,j] = concat(x[bp,i], x[bp,j])
// ---------------------------------------------------------------------------
__global__ void build_pair_kernel(const float* __restrict__ x,
                                  float* __restrict__ pair)
{
    const int idx = blockIdx.x;                 // 0..NPAIR-1
    const int tid = threadIdx.x;                // 0..255
    const int j = idx % KQ;
    const int i = (idx / KQ) % KQ;
    const int bp = idx / (KQ * KQ);
    pair[(size_t)idx * (2 * DD) + tid]      = x[(size_t)(bp * KQ + i) * DD + tid];
    pair[(size_t)idx * (2 * DD) + DD + tid] = x[(size_t)(bp * KQ + j) * DD + tid];
}

// y[bp,i,d] = x[bp,i,d] + b2[d] + (1/K) * sum_j y2[bp,i,j,d]
__global__ void mean_resid_kernel(const float* __restrict__ y2,
                                  const float* __restrict__ b2,
                                  const float* __restrict__ xin,
                                  float* __restrict__ xout)
{
    const int row = blockIdx.x;                 // 0..NTOK-1 -> (bp, i)
    const int tid = threadIdx.x;                // 0..255
    const size_t base = (size_t)row * KQ * DD;  // y2 rows are (bp*KQ+i)*KQ + j
    float acc = 0.f;
    for (int j = 0; j < KQ; ++j) acc += y2[base + (size_t)j * DD + tid];
    xout[(size_t)row * DD + tid] =
        xin[(size_t)row * DD + tid] + b2[tid] + acc * (1.f / KQ);
}

// ---------------------------------------------------------------------------
// Host orchestration
// ---------------------------------------------------------------------------
extern "C" void kernel_launch(void* const* d_in, const int* in_sizes, int n_in,
                              void* d_out, int out_size, void* d_ws, size_t ws_size,
                              hipStream_t stream)
{
    const float* memory   = (const float*)d_in[0];
    const float* person_q = (const float*)d_in[1];
    const float* key_q    = (const float*)d_in[2];
    const float* sa_in_w  = (const float*)d_in[3];
    const float* sa_in_b  = (const float*)d_in[4];
    const float* sa_out_w = (const float*)d_in[5];
    const float* sa_out_b = (const float*)d_in[6];
    const float* ca_in_w  = (const float*)d_in[7];
    const float* ca_in_b  = (const float*)d_in[8];
    const float* ca_out_w = (const float*)d_in[9];
    const float* ca_out_b = (const float*)d_in[10];
    const float* ff1_w    = (const float*)d_in[11];
    const float* ff1_b    = (const float*)d_in[12];
    const float* ff2_w    = (const float*)d_in[13];
    const float* ff2_b    = (const float*)d_in[14];
    const float* ln1_w    = (const float*)d_in[15];
    const float* ln1_b    = (const float*)d_in[16];
    const float* ln2_w    = (const float*)d_in[17];
    const float* ln2_b    = (const float*)d_in[18];
    const float* ln3_w    = (const float*)d_in[19];
    const float* ln3_b    = (const float*)d_in[20];
    const float* r1_w1    = (const float*)d_in[21];
    const float* r1_b1    = (const float*)d_in[22];
    const float* r1_w2    = (const float*)d_in[23];
    const float* r1_b2    = (const float*)d_in[24];
    const float* r2_w1    = (const float*)d_in[25];
    const float* r2_b1    = (const float*)d_in[26];
    const float* r2_w2    = (const float*)d_in[27];
    const float* r2_b2    = (const float*)d_in[28];
    const float* fc_w     = (const float*)d_in[29];
    const float* fc_b     = (const float*)d_in[30];

    float* ws = (float*)d_ws;
    size_t off = 0;
    auto alloc = [&](size_t n) { float* p = ws + off; off += n; return p; };
    float* xA     = alloc((size_t)NTOK * DD);
    float* xB     = alloc((size_t)NTOK * DD);
    float* qkv    = alloc((size_t)NTOK * 3 * DD);
    float* attn_o = alloc((size_t)NTOK * DD);
    float* qproj  = alloc((size_t)NTOK * DD);
    float* tmp    = alloc((size_t)NTOK * FFD);
    float* fc_pad = alloc((size_t)64 * DD);      // fc_w padded to 64 rows
    float* mk     = alloc((size_t)NMEM * DD);
    float* mv     = alloc((size_t)NMEM * DD);
    float* pair   = alloc((size_t)NPAIR * 2 * DD);
    float* hbuf   = alloc((size_t)NPAIR * HIDD);
    float* y2     = alloc((size_t)NPAIR * DD);
    (void)ws_size; (void)in_sizes; (void)n_in; (void)out_size;

    // Stage fc_w into padded buffer (rows 3..63 are don't-care: their output
    // columns are never stored).  Async d2d copy is graph-capture safe.
    hipMemcpyAsync(fc_pad, fc_w, (size_t)3 * DD * sizeof(float),
                   hipMemcpyDeviceToDevice, stream);

    init_x_kernel<<<NTOK, 256, 0, stream>>>(person_q, key_q, xA);

    for (int l = 0; l < LL; ++l) {
        const float* sawl  = sa_in_w  + (size_t)l * 3 * DD * DD;
        const float* sabl  = sa_in_b  + (size_t)l * 3 * DD;
        const float* sowl  = sa_out_w + (size_t)l * DD * DD;
        const float* sobl  = sa_out_b + (size_t)l * DD;
        const float* cawl  = ca_in_w  + (size_t)l * 3 * DD * DD;
        const float* cabl  = ca_in_b  + (size_t)l * 3 * DD;
        const float* cowl  = ca_out_w + (size_t)l * DD * DD;
        const float* cobl  = ca_out_b + (size_t)l * DD;
        const float* f1wl  = ff1_w + (size_t)l * FFD * DD;
        const float* f1bl  = ff1_b + (size_t)l * FFD;
        const float* f2wl  = ff2_w + (size_t)l * DD * FFD;
        const float* f2bl  = ff2_b + (size_t)l * DD;

        // --- self attention ---
        gemm_wmma_kernel<false><<<gemm_blocks(NTOK, 3 * DD), 256, 0, stream>>>(
            xA, sawl, sabl, qkv, NTOK, 3 * DD, 3 * DD, DD);
        self_attn_kernel<<<BB * PP, 256, 0, stream>>>(qkv, attn_o);
        gemm_wmma_kernel<false><<<gemm_blocks(NTOK, DD), 256, 0, stream>>>(
            attn_o, sowl, sobl, tmp, NTOK, DD, DD, DD);
        ln_resid_kernel<<<NTOK, 256, 0, stream>>>(
            xA, tmp, ln1_w + (size_t)l * DD, ln1_b + (size_t)l * DD, xA);

        // --- cross attention ---
        gemm_wmma_kernel<false><<<gemm_blocks(NTOK, DD), 256, 0, stream>>>(
            xA, cawl, cabl, qproj, NTOK, DD, DD, DD);
        gemm_wmma_kernel<false><<<gemm_blocks(NMEM, DD), 256, 0, stream>>>(
            memory, cawl + (size_t)DD * DD, cabl + DD, mk, NMEM, DD, DD, DD);
        gemm_wmma_kernel<false><<<gemm_blocks(NMEM, DD), 256, 0, stream>>>(
            memory, cawl + (size_t)2 * DD * DD, cabl + 2 * DD, mv, NMEM, DD, DD, DD);
        cross_attn_kernel<<<BB * NHH * PP * KQ, 256, 0, stream>>>(qproj, mk, mv, attn_o);
        gemm_wmma_kernel<false><<<gemm_blocks(NTOK, DD), 256, 0, stream>>>(
            attn_o, cowl, cobl, tmp, NTOK, DD, DD, DD);
        ln_resid_kernel<<<NTOK, 256, 0, stream>>>(
            xA, tmp, ln2_w + (size_t)l * DD, ln2_b + (size_t)l * DD, xA);

        // --- feed forward ---
        gemm_wmma_kernel<true><<<gemm_blocks(NTOK, FFD), 256, 0, stream>>>(
            xA, f1wl, f1bl, tmp, NTOK, FFD, FFD, DD);
        gemm_wmma_kernel<false><<<gemm_blocks(NTOK, DD), 256, 0, stream>>>(
            tmp, f2wl, f2bl, attn_o, NTOK, DD, DD, FFD);
        ln_resid_kernel<<<NTOK, 256, 0, stream>>>(
            xA, attn_o, ln3_w + (size_t)l * DD, ln3_b + (size_t)l * DD, xA);
    }

    // --- refine 1: xA -> xB ---
    build_pair_kernel<<<NPAIR, 256, 0, stream>>>(xA, pair);
    gemm_wmma_kernel<true><<<gemm_blocks(NPAIR, HIDD), 256, 0, stream>>>(
        pair, r1_w1, r1_b1, hbuf, NPAIR, HIDD, HIDD, 2 * DD);
    gemm_wmma_kernel<false><<<gemm_blocks(NPAIR, DD), 256, 0, stream>>>(
        hbuf, r1_w2, nullptr, y2, NPAIR, DD, DD, HIDD);
    mean_resid_kernel<<<NTOK, 256, 0, stream>>>(y2, r1_b2, xA, xB);

    // --- refine 2: xB -> xA ---
    build_pair_kernel<<<NPAIR, 256, 0, stream>>>(xB, pair);
    gemm_wmma_kernel<true><<<gemm_blocks(NPAIR, HIDD), 256, 0, stream>>>(
        pair, r2_w1, r2_b1, hbuf, NPAIR, HIDD, HIDD, 2 * DD);
    gemm_wmma_kernel<false><<<gemm_blocks(NPAIR, DD), 256, 0, stream>>>(
        hbuf, r2_w2, nullptr, y2, NPAIR, DD, DD, HIDD);
    mean_resid_kernel<<<NTOK, 256, 0, stream>>>(y2, r2_b2, xB, xA);

    // --- final fc: (NTOK,256) @ (3,256)^T -> d_out  (padded weights, Nc=3) ---
    gemm_wmma_kernel<false><<<gemm_blocks(NTOK, 64), 256, 0, stream>>>(
        xA, fc_pad, fc_b, (float*)d_out, NTOK, 64, 3, DD);
}